// GCN_90701119357321
// MI455X (gfx1250) — compile-verified
//
#include <hip/hip_runtime.h>
#include <hip/hip_bf16.h>

typedef __attribute__((ext_vector_type(2))) float v2f;
typedef __attribute__((ext_vector_type(8))) float v8f;

#define D 128  // feature dim for all layers

// ---------------------------------------------------------------------------
// Degree accumulation: out_deg[src[e]] += 1, in_deg[dst[e]] += 1
// ---------------------------------------------------------------------------
__global__ void gcn_deg_kernel(const int* __restrict__ src,
                               const int* __restrict__ dst,
                               float* __restrict__ out_deg,
                               float* __restrict__ in_deg, int E) {
  int e = blockIdx.x * blockDim.x + threadIdx.x;
  if (e < E) {
    unsafeAtomicAdd(&out_deg[src[e]], 1.0f);
    unsafeAtomicAdd(&in_deg[dst[e]], 1.0f);
  }
}

// deg -> clip(deg,1)^-0.5 in place
__global__ void gcn_norm_kernel(float* __restrict__ out_deg,
                                float* __restrict__ in_deg, int N) {
  int i = blockIdx.x * blockDim.x + threadIdx.x;
  if (i < N) {
    out_deg[i] = rsqrtf(fmaxf(out_deg[i], 1.0f));
    in_deg[i]  = rsqrtf(fmaxf(in_deg[i], 1.0f));
  }
}

// ---------------------------------------------------------------------------
// Edge scatter: agg[dst[e], :] += h[src[e], :] * norm_src[src[e]]
// One wave32 per edge; each lane moves 4 contiguous floats (32*4 = 128).
// ---------------------------------------------------------------------------
__global__ void __launch_bounds__(256)
gcn_scatter_kernel(const float* __restrict__ h,
                   const int* __restrict__ src,
                   const int* __restrict__ dst,
                   const float* __restrict__ norm_src,
                   float* __restrict__ agg, int E) {
  int gid  = blockIdx.x * blockDim.x + threadIdx.x;
  int edge = gid >> 5;
  int lane = threadIdx.x & 31;
  if (edge >= E) return;
  int s = src[edge];
  int d = dst[edge];
  float ns = norm_src[s];
  float4 v = *(const float4*)(h + (size_t)s * D + lane * 4);
  float* ap = agg + (size_t)d * D + lane * 4;
  unsafeAtomicAdd(ap + 0, v.x * ns);
  unsafeAtomicAdd(ap + 1, v.y * ns);
  unsafeAtomicAdd(ap + 2, v.z * ns);
  unsafeAtomicAdd(ap + 3, v.w * ns);
}

// ---------------------------------------------------------------------------
// out[i,:] = gelu_exact( (agg[i,:] * norm_dst[i]) @ W + bias )
// Block = 256 threads = 8 waves. Block owns a 16-row tile; A (16x128, scaled)
// staged in LDS once per block. Wave w computes the 16x16 tile at columns
// [16w, 16w+16) using v_wmma_f32_16x16x4_f32 over 32 K-steps.
//
// fp32 WMMA layouts (CDNA5 ISA):
//   A 16x4:  lanes 0-15 -> M=lane, VGPR0=K0, VGPR1=K1; lanes 16-31 -> K2,K3
//   B 4x16:  VGPR0: lanes0-15 row K0, lanes16-31 row K2; VGPR1: K1 / K3
//   C 16x16: VGPR v: lanes0-15 -> M=v, N=lane; lanes16-31 -> M=v+8
// ---------------------------------------------------------------------------
__global__ void __launch_bounds__(256)
gcn_gemm_gelu_kernel(const float* __restrict__ agg,
                     const float* __restrict__ Wm,
                     const float* __restrict__ bias,
                     const float* __restrict__ norm_dst,
                     float* __restrict__ out, int N) {
  __shared__ float ldsA[16][132];  // 132-pad: banks (4m + c) % 64 conflict-free

  const int row0 = blockIdx.x * 16;
  const int tid  = threadIdx.x;

  // Cooperative load of the 16x128 A tile, scaled by norm_dst. 256 threads *
  // 8 floats = 2048 = 16*128. Each rep: one float4 per thread, coalesced.
  {
    int idx = tid * 4;
#pragma unroll
    for (int rep = 0; rep < 2; ++rep) {
      int r = idx >> 7;        // 0..15
      int c = idx & 127;       // multiple of 4
      int row = row0 + r;
      float4 v = make_float4(0.f, 0.f, 0.f, 0.f);
      float nd = 0.f;
      if (row < N) {
        nd = norm_dst[row];
        v = *(const float4*)(agg + (size_t)row * D + c);
      }
      ldsA[r][c + 0] = v.x * nd;
      ldsA[r][c + 1] = v.y * nd;
      ldsA[r][c + 2] = v.z * nd;
      ldsA[r][c + 3] = v.w * nd;
      idx += 1024;
    }
  }
  __syncthreads();

  const int wave = tid >> 5;        // 0..7 -> column tile
  const int lane = tid & 31;
  const int m    = lane & 15;       // A row within tile / output N index
  const int hi   = lane >> 4;       // half-wave selector
  const int col  = wave * 16 + m;   // output column

  v8f acc = {};
#pragma unroll 4
  for (int k = 0; k < D; k += 4) {
    v2f a, b;
    a.x = ldsA[m][k + 2 * hi];
    a.y = ldsA[m][k + 2 * hi + 1];
    b.x = Wm[(size_t)(k + 2 * hi) * D + col];
    b.y = Wm[(size_t)(k + 2 * hi + 1) * D + col];
    acc = __builtin_amdgcn_wmma_f32_16x16x4_f32(
        /*neg_a=*/false, a, /*neg_b=*/false, b,
        /*c_mod=*/(short)0, acc, /*reuse_a=*/false, /*reuse_b=*/false);
  }

  // Epilogue: bias + exact GELU, store. VGPR v holds row (v + 8*hi).
  const float bb = bias[col];
#pragma unroll
  for (int v = 0; v < 8; ++v) {
    int row = row0 + v + 8 * hi;
    if (row < N) {
      float x = acc[v] + bb;
      float g = 0.5f * x * (1.0f + erff(x * 0.70710678118654752f));
      out[(size_t)row * D + col] = g;
    }
  }
}

// ---------------------------------------------------------------------------
// Host-side orchestration (graph-capture safe: only async ops on `stream`)
// ---------------------------------------------------------------------------
extern "C" void kernel_launch(void* const* d_in, const int* in_sizes, int n_in,
                              void* d_out, int out_size, void* d_ws, size_t ws_size,
                              hipStream_t stream) {
  const float* features = (const float*)d_in[0];
  const int*   src      = (const int*)d_in[1];
  const int*   dst      = (const int*)d_in[2];
  const float* Ws[3]    = {(const float*)d_in[3], (const float*)d_in[5], (const float*)d_in[7]};
  const float* bs[3]    = {(const float*)d_in[4], (const float*)d_in[6], (const float*)d_in[8]};

  const int N = in_sizes[0] / D;
  const int E = in_sizes[1];

  // Workspace layout (floats): norm_src[N] | norm_dst[N] | h[N*D] | agg[N*D]
  float* ws       = (float*)d_ws;
  float* norm_src = ws;
  float* norm_dst = ws + N;
  float* hbuf     = ws + 2 * (size_t)N;
  float* aggbuf   = hbuf + (size_t)N * D;

  // Degrees -> norms (recomputed every call; deterministic)
  hipMemsetAsync(norm_src, 0, 2 * (size_t)N * sizeof(float), stream);
  gcn_deg_kernel<<<(E + 255) / 256, 256, 0, stream>>>(src, dst, norm_src, norm_dst, E);
  gcn_norm_kernel<<<(N + 255) / 256, 256, 0, stream>>>(norm_src, norm_dst, N);

  const int scatter_blocks = (E + 7) / 8;   // 8 edges (waves) per 256-thread block
  const int gemm_blocks    = (N + 15) / 16;

  const float* hin = features;
  for (int l = 0; l < 3; ++l) {
    hipMemsetAsync(aggbuf, 0, (size_t)N * D * sizeof(float), stream);
    gcn_scatter_kernel<<<scatter_blocks, 256, 0, stream>>>(hin, src, dst,
                                                           norm_src, aggbuf, E);
    float* hout = (l == 2) ? (float*)d_out : hbuf;
    gcn_gemm_gelu_kernel<<<gemm_blocks, 256, 0, stream>>>(aggbuf, Ws[l], bs[l],
                                                          norm_dst, hout, N);
    hin = hbuf;
  }
}